// ScaledDotProductAttention_59493886984760
// MI455X (gfx1250) — compile-verified
//
#include <hip/hip_runtime.h>
#include <cstdint>
#include <cstddef>

// ---------------------------------------------------------------------------
// Flash-attention style fused SDPA for MI455X (gfx1250, wave32, WMMA).
//
// Roofline: must move ~1.16 GB (res_att read + mask read + scores write +
// context write) -> ~50us at 23.3 TB/s HBM. Compute is ~17 GFLOP f32, hidden
// behind memory even with the f32 16x16x4 WMMA. Single pass: scores written
// once, never re-read; softmax is online (flash) with lane-local row stats.
// ---------------------------------------------------------------------------

typedef __attribute__((ext_vector_type(2))) float v2f;
typedef __attribute__((ext_vector_type(8))) float v8f;

// D = A(16x4 f32) * B(4x16 f32) + C(16x16 f32)
// 8-arg pattern: (neg_a, A, neg_b, B, c_mod, C, reuse_a, reuse_b)
#define WMMA_F32x4(A, B, C) \
  __builtin_amdgcn_wmma_f32_16x16x4_f32(false, (A), false, (B), (short)0, (C), false, false)

namespace {
constexpr int Bb = 4, Hh = 8, Ll = 2048, DK = 32, DV = 32;
constexpr int QT = 16, KT = 16;              // 16x16 WMMA tiles
constexpr size_t CTX_ELEMS = (size_t)Bb * Hh * Ll * DV;   // context comes first in d_out
}

__global__ __launch_bounds__(32)
void sdpa_flash_wmma(const float* __restrict__ Q,
                     const float* __restrict__ K,
                     const float* __restrict__ V,
                     const unsigned char* __restrict__ mask,
                     const float* __restrict__ res,
                     float* __restrict__ ctx_out,
                     float* __restrict__ scores_out)
{
  __shared__ float plds[QT * KT];          // P tile staging (C-layout -> A-layout)

  const int lane = threadIdx.x & 31;
  const int lh = lane >> 4;                // which 16-lane half
  const int ln = lane & 15;                // lane within half

  const int qt = blockIdx.x % (Ll / QT);
  const int bh = blockIdx.x / (Ll / QT);   // fused (b,h)
  const int q0 = qt * QT;

  const float* Qb = Q + (size_t)bh * Ll * DK;
  const float* Kb = K + (size_t)bh * Ll * DK;
  const float* Vb = V + (size_t)bh * Ll * DV;
  const unsigned char* Mb = mask + (size_t)bh * Ll * Ll;
  const float* Rb = res + (size_t)bh * Ll * Ll;
  float* Sb = scores_out + (size_t)bh * Ll * Ll;
  float* Cb = ctx_out + (size_t)bh * Ll * DV;

  // --- preload Q tile as 8 A-operand slices (16x4 each) -------------------
  // ISA A layout (32-bit 16x4): VGPR0 = K {0|2}, VGPR1 = K {1|3} by lane half.
  v2f qa[8];
#pragma unroll
  for (int kk = 0; kk < 8; ++kk) {
    const float* p = Qb + (size_t)(q0 + ln) * DK + 4 * kk + 2 * lh;
    qa[kk].x = p[0];
    qa[kk].y = p[1];
  }

  // Online-softmax state: rows r+8*lh live on element r for this lane's half,
  // so all row statistics are lane-local after a 16-lane xor-reduction.
  float mrow[8], lrow[8];
  v8f O0 = {}, O1 = {};                    // context accumulator, DV split 16+16
#pragma unroll
  for (int r = 0; r < 8; ++r) { mrow[r] = -3.0e38f; lrow[r] = 0.0f; }

  const float scale = 0.17677669529663687f;   // 1/sqrt(32)

  for (int j = 0; j < Ll / KT; ++j) {
    const int k0 = j * KT;

    // --- S = Q * K^T : 8 chained f32 WMMAs over DK=32 ---------------------
    v8f S = {};
#pragma unroll
    for (int kk = 0; kk < 8; ++kk) {
      v2f kb;   // B operand (4x16 of K^T) == same addressing as A on K
      const float* p = Kb + (size_t)(k0 + ln) * DK + 4 * kk + 2 * lh;
      kb.x = p[0];
      kb.y = p[1];
      S = WMMA_F32x4(qa[kk], kb, S);
    }

    // --- scale + res_att + mask, emit scores, online softmax update ------
    float pv[8];
#pragma unroll
    for (int r = 0; r < 8; ++r) {
      const int m = r + 8 * lh;                       // row within tile
      const size_t idx = (size_t)(q0 + m) * Ll + (size_t)(k0 + ln);
      float sc = S[r] * scale + Rb[idx];
      if (Mb[idx]) sc = -1e9f;
      Sb[idx] = sc;                                   // second output: scores

      // row max over the 16 lanes of this half (xor masks stay in-half)
      float tm = sc;
      tm = fmaxf(tm, __shfl_xor(tm, 1, 32));
      tm = fmaxf(tm, __shfl_xor(tm, 2, 32));
      tm = fmaxf(tm, __shfl_xor(tm, 4, 32));
      tm = fmaxf(tm, __shfl_xor(tm, 8, 32));

      const float mn = fmaxf(mrow[r], tm);
      const float alpha = __expf(mrow[r] - mn);       // rescale factor
      const float p = __expf(sc - mn);

      float ps = p;                                   // row sum of P
      ps += __shfl_xor(ps, 1, 32);
      ps += __shfl_xor(ps, 2, 32);
      ps += __shfl_xor(ps, 4, 32);
      ps += __shfl_xor(ps, 8, 32);

      lrow[r] = lrow[r] * alpha + ps;
      mrow[r] = mn;
      O0[r] *= alpha;
      O1[r] *= alpha;
      pv[r] = p;
    }

    // --- P: C/D layout -> A layout via LDS --------------------------------
    __syncthreads();                                  // guard LDS reuse
#pragma unroll
    for (int r = 0; r < 8; ++r)
      plds[(r + 8 * lh) * KT + ln] = pv[r];
    __syncthreads();

    // --- O += P(16x16) * V(16x32) : 4 WMMA slices x 2 DV chunks -----------
#pragma unroll
    for (int kk = 0; kk < 4; ++kk) {
      v2f pa;
      pa.x = plds[ln * KT + 4 * kk + 2 * lh];
      pa.y = plds[ln * KT + 4 * kk + 2 * lh + 1];

      const float* v0 = Vb + (size_t)(k0 + 4 * kk + 2 * lh) * DV;  // key row
      const float* v1 = v0 + DV;                                    // next key
      v2f vb0, vb1;
      vb0.x = v0[ln];      vb0.y = v1[ln];        // DV columns 0..15
      vb1.x = v0[16 + ln]; vb1.y = v1[16 + ln];   // DV columns 16..31
      O0 = WMMA_F32x4(pa, vb0, O0);
      O1 = WMMA_F32x4(pa, vb1, O1);
    }
  }

  // --- finalize: context = O / l -------------------------------------------
#pragma unroll
  for (int r = 0; r < 8; ++r) {
    const int m = r + 8 * lh;
    const float inv = 1.0f / lrow[r];
    Cb[(size_t)(q0 + m) * DV + ln]      = O0[r] * inv;
    Cb[(size_t)(q0 + m) * DV + 16 + ln] = O1[r] * inv;
  }
}

extern "C" void kernel_launch(void* const* d_in, const int* in_sizes, int n_in,
                              void* d_out, int out_size, void* d_ws, size_t ws_size,
                              hipStream_t stream) {
  (void)in_sizes; (void)n_in; (void)out_size; (void)d_ws; (void)ws_size;

  const float* Q = (const float*)d_in[0];
  const float* K = (const float*)d_in[1];
  const float* V = (const float*)d_in[2];
  const unsigned char* mask = (const unsigned char*)d_in[3];  // jnp bool = 1 byte
  const float* res = (const float*)d_in[4];

  float* out = (float*)d_out;
  float* ctx = out;                 // [B,H,L,DV] first (return order)
  float* scores = out + CTX_ELEMS;  // [B,H,L,L] second

  const int nblocks = Bb * Hh * (Ll / QT);   // 4096 q-tiles
  sdpa_flash_wmma<<<dim3(nblocks), dim3(32), 0, stream>>>(
      Q, K, V, mask, res, ctx, scores);
}